// GELU62_17566416240710
// MI455X (gfx1250) — compile-verified
//
#include <hip/hip_runtime.h>
#include <cmath>

// Problem constants (match reference)
constexpr int Bn = 8;
constexpr int Tn = 2048;
constexpr int Dn = 2048;

// Tiling
constexpr int CH     = 128;            // channels per workgroup (== blockDim.x)
constexpr int TC     = 128;            // t-steps per LDS tile
constexpr int WAVES  = CH / 32;        // 4 waves per WG
constexpr int ROWSPW = TC / WAVES;     // 32 async rows issued per wave per tile
constexpr int CHUNKS = Dn / CH;        // 16 channel-chunks per batch
constexpr int NTILE  = Tn / TC;        // 16 tiles along T
constexpr int NPART  = CHUNKS * WAVES; // 64 partial sums per (b,t)
constexpr int BT     = Bn * Tn;        // 16384

static_assert(TC == CH, "smean staging assumes one inv_mean element per thread");

// CDNA5 hardware tanh (V_TANH_F32) when the toolchain declares it.
#if defined(__has_builtin)
#if __has_builtin(__builtin_amdgcn_tanhf)
#define FAST_TANH(x) __builtin_amdgcn_tanhf(x)
#endif
#endif
#ifndef FAST_TANH
#define FAST_TANH(x) tanhf(x)
#endif

__device__ __forceinline__ void async_row_b128(unsigned lds_addr, unsigned goff,
                                               const float* base) {
  // VDST = per-lane LDS byte address, VADDR = per-lane 32-bit byte offset,
  // SADDR = 64-bit base (GVS mode). Tracked by ASYNCcnt.
  asm volatile("global_load_async_to_lds_b128 %0, %1, %2"
               :: "v"(lds_addr), "v"(goff), "s"(base) : "memory");
}

__device__ __forceinline__ void wait_async_le32() {
  asm volatile("s_wait_asynccnt 32" ::: "memory");
}
__device__ __forceinline__ void wait_async_0() {
  asm volatile("s_wait_asynccnt 0" ::: "memory");
}

// Shared body. WRITE_OUT=false: emit per-wave partial sums of r (pass 1).
// WRITE_OUT=true: write gelu(x) * r * inv_mean (pass 2).
template <bool WRITE_OUT>
__device__ __forceinline__ void scan_body(
    const float* __restrict__ x, const float* __restrict__ ema,
    const float* __restrict__ logitU, const float* __restrict__ logtau,
    float* __restrict__ partials, const float* __restrict__ inv_mean,
    float* __restrict__ out) {
  __shared__ float tile[2][TC * CH];  // 128 KB double buffer
  __shared__ float smean[TC];

  const int b     = blockIdx.x / CHUNKS;
  const int chunk = blockIdx.x % CHUNKS;
  const int c0    = chunk * CH;
  const int tid   = threadIdx.x;
  const int lane  = tid & 31;
  // Wave index is uniform across the wave; tell the compiler so the async
  // issue loop is uniform (no exec-mask bookkeeping, fully unrollable).
  const int wave  = __builtin_amdgcn_readfirstlane(tid >> 5);

  // Scalar hyper-params (reference: sigmoid / softplus, tau>=0.5)
  const float U_val = 1.0f / (1.0f + expf(-logitU[0]));
  float tau = log1pf(expf(logtau[0]));
  tau = fmaxf(tau, 0.5f);
  const float rec_rate = 1.0f - expf(-1.0f / tau);

  float r = ema[c0 + tid];  // r0 = broadcast ema_resource

  const unsigned tbase0 = (unsigned)(uintptr_t)(&tile[0][0]);
  const unsigned tbase1 = (unsigned)(uintptr_t)(&tile[1][0]);
  const unsigned rowB   = CH * 4;  // 512 B per tile row

  // One async b128 row per wave-instruction: lane l covers channels c0+4l..+3.
  // Rows are distributed round-robin across the 4 waves; uniform trip count.
  auto issue_tile = [&](int ti, int buf) {
    const unsigned lds0 = (buf ? tbase1 : tbase0) + (unsigned)(wave * rowB)
                          + (unsigned)(lane * 16);
    const unsigned g0   = ((unsigned)((b * Tn + ti * TC + wave) * Dn + c0)) * 4u
                          + (unsigned)(lane * 16);
#pragma unroll
    for (int k = 0; k < ROWSPW; ++k)
      async_row_b128(lds0 + (unsigned)(k * WAVES) * rowB,
                     g0 + (unsigned)(k * WAVES) * (unsigned)(Dn * 4), x);
  };

  issue_tile(0, 0);  // prologue

  const float kC  = 0.7978845608028654f;  // sqrt(2/pi)
  const float kCA = kC * 0.044715f;

  for (int ti = 0; ti < NTILE; ++ti) {
    const int cur = ti & 1;
    const bool more = (ti + 1) < NTILE;
    if (more) issue_tile(ti + 1, cur ^ 1);

    if constexpr (WRITE_OUT) {  // stage inv_mean row for this tile (TC==CH)
      smean[tid] = inv_mean[b * Tn + ti * TC + tid];
    }

    if (more) wait_async_le32(); else wait_async_0();
    __syncthreads();  // tile `ti` + smean visible to all waves

    const int t0 = ti * TC;
#pragma unroll 4
    for (int tl = 0; tl < TC; ++tl) {
      const float v = tile[cur][tl * CH + tid];
      // tanh-approx GELU in FMA form: u = v*(kC + kC*kA*v^2)
      const float v2 = v * v;
      const float u  = v * fmaf(kCA, v2, kC);
      const float th = FAST_TANH(u);
      const float hv = 0.5f * v;
      const float g  = fmaf(hv, th, hv);
      // firing; |g| folds into the tanh source modifier
      const float f = FAST_TANH(fabsf(g));
      // resource-depression step (emit r BEFORE update)
      const float r_pre = r;
      const float used  = fminf(U_val * r * f, 0.99f * r);
      r = (r - used) + fmaf(-rec_rate, r, rec_rate);  // + (1-r)*rec
      r = fminf(fmaxf(r, 0.01f), 1.0f);

      if constexpr (WRITE_OUT) {
        out[(b * Tn + t0 + tl) * Dn + c0 + tid] = g * r_pre * smean[tl];
      } else {
        // deterministic wave-level sum of r_pre over 32 channels
        float s = r_pre;
#pragma unroll
        for (int o = 16; o > 0; o >>= 1) s += __shfl_xor(s, o, 32);
        if (lane == 0)
          partials[(chunk * WAVES + wave) * BT + b * Tn + t0 + tl] = s;
      }
    }
    __syncthreads();  // seal reads before buffer/smean reuse
  }
}

// Defined first so the disassembly snippet shows the hot loop.
__global__ __launch_bounds__(CH) void scan_sums_kernel(
    const float* __restrict__ x, const float* __restrict__ ema,
    const float* __restrict__ logitU, const float* __restrict__ logtau,
    float* __restrict__ partials) {
  scan_body<false>(x, ema, logitU, logtau, partials, nullptr, nullptr);
}

__global__ __launch_bounds__(CH) void scan_out_kernel(
    const float* __restrict__ x, const float* __restrict__ ema,
    const float* __restrict__ logitU, const float* __restrict__ logtau,
    const float* __restrict__ inv_mean, float* __restrict__ out) {
  scan_body<true>(x, ema, logitU, logtau, nullptr, inv_mean, out);
}

// Fixed-order reduction of 64 partials -> 1/(mean_D + eps). Deterministic.
__global__ __launch_bounds__(256) void reduce_kernel(
    const float* __restrict__ partials, float* __restrict__ inv_mean) {
  const int i = blockIdx.x * blockDim.x + threadIdx.x;
  if (i >= BT) return;
  float s = 0.0f;
#pragma unroll
  for (int p = 0; p < NPART; ++p) s += partials[p * BT + i];
  inv_mean[i] = 1.0f / (s * (1.0f / (float)Dn) + 1e-5f);
}

extern "C" void kernel_launch(void* const* d_in, const int* in_sizes, int n_in,
                              void* d_out, int out_size, void* d_ws, size_t ws_size,
                              hipStream_t stream) {
  const float* x      = (const float*)d_in[0];
  const float* ema    = (const float*)d_in[1];
  const float* logitU = (const float*)d_in[2];
  const float* logtau = (const float*)d_in[3];
  // d_in[4] (logit_decay) is unused by the reference forward pass.
  float* out = (float*)d_out;

  float* partials = (float*)d_ws;                  // NPART*BT floats = 4 MiB
  float* inv_mean = partials + (size_t)NPART * BT; // BT floats = 64 KiB

  dim3 grid(Bn * CHUNKS), block(CH);
  scan_sums_kernel<<<grid, block, 0, stream>>>(x, ema, logitU, logtau, partials);
  reduce_kernel<<<BT / 256, 256, 0, stream>>>(partials, inv_mean);
  scan_out_kernel<<<grid, block, 0, stream>>>(x, ema, logitU, logtau, inv_mean, out);
}